// RAE_60112362275057
// MI455X (gfx1250) — compile-verified
//
#include <hip/hip_runtime.h>
#include <math.h>

// ---------------- problem constants ----------------
#define T_STEPS 64
#define BATCH   256
#define XD      128
#define HD      1024
#define SKIP    5

#define GH_SPLIT   8     // K-split of the 256x3072x1024 recurrent GEMM
#define PROJ_SPLIT 8     // K-split of the 256x128x1024 projection GEMM

#define BK   16          // K-depth of one LDS tile buffer
#define AST  20          // padded LDS row stride (floats): 80B, 16B-aligned,
#define BST  20          // and 20*r mod 64 hits distinct banks for r=0..15

typedef __attribute__((ext_vector_type(2))) float v2f;
typedef __attribute__((ext_vector_type(8))) float v8f;

// =====================================================================
// fp32 WMMA GEMM, LDS-staged with CDNA5 async load-to-LDS double
// buffering (ASYNCcnt + workgroup barrier pipeline).
//
// Geometry (fixed): block = 256 threads = 8 waves.
//   block tile = 256 rows (8 waves x MT=2 x 16) x 64 cols (NT=4 x 16)
//   -> the B tile (weights) is shared by all 8 waves through LDS,
//      cutting L2 weight traffic 8x vs per-wave fragment loads.
// Grid = S (K-split) x MB (row blocks) x nBlocks (col blocks).
//   D_s[row][n] = sum_{k in chunk s} sA*A[rowmap(row)*lda+k] * W[n*ldw+k]
// rowmap: row = mb*256 + r -> A offset r*lda + mb*chunkStride
//   (used to batch 8 encoder timesteps into one launch)
//
// WMMA f32 16x16x4 fragment mapping (CDNA5 ISA):
//   A (16x4): lane m=lane&15, half=lane>>4: {A[m][k+2h], A[m][k+2h+1]}
//   B (4x16): lane n=lane&15, half h: {B[k+2h][n], B[k+2h+1][n]},
//             B[k][n] = W[n][k]
//   C (16x16): c[r] = D[m0 + r + 8*half][n0 + (lane&15)]
// =====================================================================
__global__ __launch_bounds__(256) void rae_gemm_wmma_lds(
    const float* __restrict__ A, float sA, int lda, int chunkStride,
    const float* __restrict__ W, int ldw, int kChunk, int nBlocks, int MB,
    const float* __restrict__ bias,
    float* __restrict__ D, int ldd, size_t partStride)
{
    __shared__ __align__(16) float lA[2][256 * AST];   // 2 x 20KB
    __shared__ __align__(16) float lB[2][64 * BST];    // 2 x  5KB

    const int tid  = threadIdx.x;
    const int lane = tid & 31;
    const int wv   = tid >> 5;

    const int bid = blockIdx.x;
    const int s   = bid / (MB * nBlocks);
    const int rem = bid % (MB * nBlocks);
    const int mb  = rem / nBlocks;
    const int nb  = rem % nBlocks;
    const int n0  = nb * 64;
    const int kbase = s * kChunk;

    // ---- staging source addresses (per thread) ----
    // A: thread tid copies row tid's BK floats per buffer (4x b128)
    const float* gA = A + (size_t)tid * lda + (size_t)mb * chunkStride + kbase;
    // B: 4 threads per W row, 1x b128 each
    const float* gB = W + (size_t)(n0 + (tid >> 2)) * ldw + kbase + (tid & 3) * 4;

    const int G = kChunk / BK;

    auto stage = [&](int g, int p) {
        const float* sa = gA + g * BK;
        unsigned da = (unsigned)(uintptr_t)&lA[p][tid * AST];
        asm volatile(
            "global_load_async_to_lds_b128 %0, %1, off\n\t"
            "global_load_async_to_lds_b128 %0, %1, off offset:16\n\t"
            "global_load_async_to_lds_b128 %0, %1, off offset:32\n\t"
            "global_load_async_to_lds_b128 %0, %1, off offset:48"
            :: "v"(da), "v"(sa) : "memory");
        const float* sb = gB + g * BK;
        unsigned db = (unsigned)(uintptr_t)&lB[p][(tid >> 2) * BST + (tid & 3) * 4];
        asm volatile(
            "global_load_async_to_lds_b128 %0, %1, off"
            :: "v"(db), "v"(sb) : "memory");
    };

    const int half = lane >> 4;
    const int mr   = lane & 15;
    const int nc   = lane & 15;
    const int koff = half << 1;

    v8f c0[4] = {}, c1[4] = {};

    stage(0, 0);
    for (int g = 0; g < G; ++g) {
        asm volatile("s_wait_asynccnt 0" ::: "memory");
        __syncthreads();
        if (g + 1 < G) stage(g + 1, (g + 1) & 1);

        const float* bufA = &lA[g & 1][0];
        const float* bufB = &lB[g & 1][0];
        #pragma unroll
        for (int ks = 0; ks < BK / 4; ++ks) {
            const int kl = ks * 4 + koff;
            v2f a0 = *(const v2f*)&bufA[(wv * 32 + mr) * AST + kl];
            v2f a1 = *(const v2f*)&bufA[(wv * 32 + 16 + mr) * AST + kl];
            a0 *= sA;
            a1 *= sA;
            v2f b[4];
            #pragma unroll
            for (int j = 0; j < 4; ++j)
                b[j] = *(const v2f*)&bufB[(j * 16 + nc) * BST + kl];
            #pragma unroll
            for (int j = 0; j < 4; ++j) {
                c0[j] = __builtin_amdgcn_wmma_f32_16x16x4_f32(
                    false, a0, false, b[j], (short)0, c0[j], false, false);
                c1[j] = __builtin_amdgcn_wmma_f32_16x16x4_f32(
                    false, a1, false, b[j], (short)0, c1[j], false, false);
            }
        }
    }

    float* Dp = D + partStride * s;
    #pragma unroll
    for (int mt = 0; mt < 2; ++mt) {
        const int mbase = mb * 256 + wv * 32 + mt * 16 + (half << 3);
        v8f* cc = mt ? c1 : c0;
        #pragma unroll
        for (int r = 0; r < 8; ++r) {
            float* drow = Dp + (size_t)(mbase + r) * ldd;
            #pragma unroll
            for (int j = 0; j < 4; ++j) {
                const int n = n0 + j * 16 + nc;
                float v = cc[j][r];
                if (bias) v += bias[n];
                drow[n] = v;
            }
        }
    }
}

// =====================================================================
// GRU gate fusion + fused mask-combines for the NEXT step:
//   gh = sum_s ghp[s] + b_hh (fixed order)
//   h_new = (1-z)*n + z*(hpScale * hcomb)
//   hout  = h_new
//   hcombN = ns1*h_new + ns2*(selfSkip ? h_new : hskipN)   (next gh's A)
//   pcomb  = p1*pOrig + p2*(pSelf ? h_new : pSkip)         (decoder proj A)
// =====================================================================
__global__ __launch_bounds__(256) void rae_gru_update(
    const float* __restrict__ gi, const float* __restrict__ ghp,
    const float* __restrict__ b_hh,
    const float* __restrict__ hcomb, float hpScale,
    float ns1, float ns2, const float* __restrict__ hskipN, int selfSkip,
    float* __restrict__ hout, float* __restrict__ hcombN,
    float p1, float p2, const float* __restrict__ pOrig,
    const float* __restrict__ pSkip, int pSelf, float* __restrict__ pcomb)
{
    const int idx = blockIdx.x * blockDim.x + threadIdx.x;   // over B*HD
    const int b = idx >> 10;
    const int j = idx & (HD - 1);
    const size_t base = (size_t)b * (3 * HD) + j;
    const size_t PS = (size_t)BATCH * 3 * HD;

    const float ir = gi[base], iz = gi[base + HD], in_ = gi[base + 2 * HD];
    float hr = b_hh[j], hz = b_hh[HD + j], hn = b_hh[2 * HD + j];
    #pragma unroll
    for (int s = 0; s < GH_SPLIT; ++s) {
        const float* g = ghp + PS * s;
        hr += g[base];
        hz += g[base + HD];
        hn += g[base + 2 * HD];
    }

    const float r = 1.f / (1.f + expf(-(ir + hr)));
    const float z = 1.f / (1.f + expf(-(iz + hz)));
    const float n = tanhf(in_ + r * hn);
    const float hp = hpScale * hcomb[idx];
    const float hnew = (1.f - z) * n + z * hp;

    hout[idx] = hnew;
    const float sk = selfSkip ? hnew : hskipN[idx];
    hcombN[idx] = ns1 * hnew + ns2 * sk;
    if (pcomb) {
        const float ps = pSelf ? hnew : pSkip[idx];
        pcomb[idx] = p1 * pOrig[idx] + p2 * ps;
    }
}

// =====================================================================
// Fixed-order reduction of PROJ_SPLIT projection partials + bias,
// strided store into d_out[b][t][:]
// =====================================================================
__global__ __launch_bounds__(256) void rae_proj_reduce(
    const float* __restrict__ parts, const float* __restrict__ bias,
    float* __restrict__ out, int ldo)
{
    const int idx = blockIdx.x * blockDim.x + threadIdx.x;   // over B*XD
    const int b = idx >> 7;
    const int n = idx & (XD - 1);
    float v = bias[n];
    #pragma unroll
    for (int s = 0; s < PROJ_SPLIT; ++s)
        v += parts[(size_t)s * BATCH * XD + idx];
    out[(size_t)b * ldo + n] = v;
}

// =====================================================================
// Host side
// =====================================================================
namespace {

// numpy legacy RandomState: MT19937 with init_genrand(seed);
// randint(0,2) draws one tempered 32-bit word, takes &1 (mask==rng==1).
struct MT19937 {
    unsigned mt[624];
    int idx;
    void seed(unsigned s) {
        mt[0] = s;
        for (int i = 1; i < 624; ++i)
            mt[i] = 1812433253u * (mt[i - 1] ^ (mt[i - 1] >> 30)) + (unsigned)i;
        idx = 624;
    }
    unsigned next() {
        if (idx >= 624) {
            for (int i = 0; i < 624; ++i) {
                unsigned y = (mt[i] & 0x80000000u) | (mt[(i + 1) % 624] & 0x7fffffffu);
                unsigned v = mt[(i + 397) % 624] ^ (y >> 1);
                if (y & 1u) v ^= 0x9908b0dfu;
                mt[i] = v;
            }
            idx = 0;
        }
        unsigned y = mt[idx++];
        y ^= y >> 11;
        y ^= (y << 7)  & 0x9d2c5680u;
        y ^= (y << 15) & 0xefc60000u;
        y ^= y >> 18;
        return y;
    }
};

// h_prev skip source (python negative-index quirk), -1 => zeros
inline int skip_src(int i) {
    if (i >= 2 * SKIP) return i - 2 * SKIP;
    if (i == 3) return 1;
    if (i == 4) return 3;
    return -1;
}
// decoder projection skip source (read AFTER the append), -1 => zeros
inline int proj_skip_src(int i) {
    if (i >= 2 * SKIP) return i - 2 * SKIP;
    if (i == 2) return 0;
    if (i == 3) return 2;
    if (i == 4) return 4;
    return -1;
}

} // namespace

extern "C" void kernel_launch(void* const* d_in, const int* in_sizes, int n_in,
                              void* d_out, int out_size, void* d_ws, size_t ws_size,
                              hipStream_t stream)
{
    (void)in_sizes; (void)n_in; (void)out_size;

    const float* input    = (const float*)d_in[0];
    const float* enc_W_ih = (const float*)d_in[1];
    const float* enc_W_hh = (const float*)d_in[2];
    const float* enc_b_ih = (const float*)d_in[3];
    const float* enc_b_hh = (const float*)d_in[4];
    const float* dec_W_ih = (const float*)d_in[5];
    const float* dec_W_hh = (const float*)d_in[6];
    const float* dec_b_ih = (const float*)d_in[7];
    const float* dec_b_hh = (const float*)d_in[8];
    const float* lin_W    = (const float*)d_in[9];
    const float* lin_b    = (const float*)d_in[10];
    float* out = (float*)d_out;

    // ---- masks: replicate np.random.RandomState(0) draw-for-draw ----
    float enc_w1[T_STEPS], enc_w2[T_STEPS], dec_w1[T_STEPS], dec_w2[T_STEPS];
    {
        MT19937 rng; rng.seed(0u);
        for (int i = 0; i < T_STEPS; ++i) {
            int w1 = (int)(rng.next() & 1u);
            int w2 = (w1 == 0) ? 1 : (int)(rng.next() & 1u);
            enc_w1[i] = (float)w1; enc_w2[i] = (float)w2;
        }
        for (int i = 0; i < T_STEPS; ++i) {
            int w1 = (int)(rng.next() & 1u);
            int w2 = (w1 == 0) ? 1 : (int)(rng.next() & 1u);
            dec_w1[i] = (float)w1; dec_w2[i] = (float)w2;
        }
    }

    // ---- workspace layout (floats) ----
    const size_t HSZ = (size_t)BATCH * HD;            // 262144
    const size_t G3  = (size_t)BATCH * 3 * HD;        // 786432
    float* ring  = (float*)d_ws;                      // 16 h-state slots
    float* ghb   = ring + 16 * HSZ;                   // GH_SPLIT partial gh
    float* projb = ghb + (size_t)GH_SPLIT * G3;       // PROJ_SPLIT partial proj
    float* hcomb = projb + (size_t)PROJ_SPLIT * BATCH * XD;  // combined h_prev
    float* gib   = hcomb + HSZ;                       // gi chunk buffer
    auto slot = [&](int j) { return ring + (size_t)(j & 15) * HSZ; };

    // encoder gi chunking: 8 timesteps per launch if workspace allows
    const size_t base_f = (size_t)(gib - (float*)d_ws);
    const int CH = (ws_size >= (base_f + 8 * G3) * sizeof(float)) ? 8 : 1;

    const int ldx = T_STEPS * XD;     // row stride of [B,T,XD] tensors
    const int N3 = 3 * HD;
    const int gru_blocks  = (int)(HSZ / 256);
    const int proj_blocks = (BATCH * XD) / 256;
    const int nB3 = N3 / 64;          // 48 col-blocks for 3072-wide GEMMs

    auto launch_update = [&](const float* gi_i, const float* bhh,
                             float hpScale, int i, bool dec) {
        // next-step h_prev combine parameters
        float ns1 = 0.f, ns2 = 0.f; const float* hskipN = slot(0); int selfSkip = 0;
        if (!dec && i == T_STEPS - 1) {             // enc 63 -> dec 0: h_enc
            ns1 = 2.f * dec_w1[0];
        } else if (i < T_STEPS - 1) {
            ns1 = dec ? dec_w1[i + 1] : enc_w1[i + 1];
            int sk2 = skip_src(i + 1);
            if (sk2 >= 0) {
                ns2 = dec ? dec_w2[i + 1] : enc_w2[i + 1];
                if (sk2 == i) selfSkip = 1; else hskipN = slot(sk2);
            }
        }
        // decoder projection combine parameters
        float p1 = 0.f, p2 = 0.f;
        const float* pOrig = slot(0); const float* pSkip = slot(0);
        int pSelf = 0; float* pcomb = nullptr;
        if (dec) {
            pcomb = projb;  // reuse: proj GEMM A reads it before overwrite? no:
            pcomb = hcomb + 0; // placeholder, replaced below
        }
        (void)pcomb;
        // (actual pcomb buffer chosen by caller variant below)
        if (dec) {
            if (i == 0) { pOrig = slot(T_STEPS - 1); p1 = 2.f; }
            else        { pOrig = slot(i - 1);       p1 = 1.f; }
            int psk = proj_skip_src(i);
            if (psk >= 0) { p2 = 1.f; if (psk == i) pSelf = 1; else pSkip = slot(psk); }
        }
        rae_gru_update<<<gru_blocks, 256, 0, stream>>>(
            gi_i, ghb, bhh, hcomb, hpScale,
            ns1, ns2, hskipN, selfSkip,
            slot(i), hcomb,
            p1, p2, pOrig, pSkip, pSelf,
            dec ? (hcomb + 0, gib + (size_t)G3) : nullptr);
    };

    // pcomb lives after gi buffer region used by decoder (gi uses G3 floats)
    float* pcombBuf = gib + G3;

    // gh GEMM: A=hcomb, 3072 cols, K=1024 split 8 ways
    auto launch_gh = [&](float sA, const float* Whh) {
        rae_gemm_wmma_lds<<<GH_SPLIT * nB3, 256, 0, stream>>>(
            hcomb, sA, HD, 0,
            Whh, HD, HD / GH_SPLIT, nB3, 1,
            nullptr, ghb, N3, G3);
    };

    // ======================= Encoder =======================
    for (int i0 = 0; i0 < T_STEPS; i0 += CH) {
        // gi for CH timesteps in one launch: row = c*256 + b
        rae_gemm_wmma_lds<<<CH * nB3, 256, 0, stream>>>(
            input + (size_t)i0 * XD, 1.f, ldx, XD,
            enc_W_ih, XD, XD, nB3, CH,
            enc_b_ih, gib, N3, 0);

        for (int cstep = 0; cstep < CH; ++cstep) {
            const int i = i0 + cstep;
            launch_gh(i == 0 ? 0.f : 1.f, enc_W_hh);
            launch_update(gib + (size_t)cstep * G3, enc_b_hh,
                          i == 0 ? 0.f : 1.f, i, false);
        }
    }

    // ======================= Decoder =======================
    for (int i = 0; i < T_STEPS; ++i) {
        // gi = x_in @ W_ih^T + b_ih ; x_in = 0 at i==0 else outputs[i-1]
        const float* xp = (i == 0) ? input : (const float*)(out + (size_t)(i - 1) * XD);
        const float  sx = (i == 0) ? 0.f : 1.f;
        rae_gemm_wmma_lds<<<nB3, 256, 0, stream>>>(
            xp, sx, ldx, 0,
            dec_W_ih, XD, XD, nB3, 1,
            dec_b_ih, gib, N3, 0);

        launch_gh(1.f, dec_W_hh);
        launch_update(gib, dec_b_hh, 1.f, i, true);

        // proj partials: A = pcombBuf (written by update), 128 cols, split 8
        rae_gemm_wmma_lds<<<PROJ_SPLIT * (XD / 64), 256, 0, stream>>>(
            pcombBuf, 1.f, HD, 0,
            lin_W, HD, HD / PROJ_SPLIT, XD / 64, 1,
            nullptr, projb, XD, (size_t)BATCH * XD);

        rae_proj_reduce<<<proj_blocks, 256, 0, stream>>>(
            projb, lin_b, out + (size_t)i * XD, ldx);
    }
}